// PaCLoss_69621419868360
// MI455X (gfx1250) — compile-verified
//
#include <hip/hip_runtime.h>
#include <hip/hip_bf16.h>
#include <math.h>

typedef float v2f __attribute__((ext_vector_type(2)));
typedef float v8f __attribute__((ext_vector_type(8)));

#define DIM   256
#define NB    16
#define NK    16
#define NNEG  16384
#define S1    257      // padded LDS row stride for inversion (kills pivot-column bank conflicts)
#define LSTR  260      // padded LDS row stride for staged negatives (260%64==4 -> conflict-free)
#define ALPHA 0.2f
#define EPSF  1e-12f

// workspace layout (in floats)
#define WS_APACK 0          // 16 * 65536 : A pair-packed into WMMA B-fragment order
#define WS_PAT   1048576    // 16 * 4096  : PA pair-packed for WMMA A-fragments
#define WS_PAP   1114112    // 256
#define WS_DPOS  1114368    // 256
#define WS_MALL  1114624    // 256 (u32 bits, float-min accumulators)
#define WS_MSH   1114880    // 256 (u32 bits)

__global__ void init_min_kernel(unsigned int* mall, unsigned int* msh) {
    int t = threadIdx.x;
    if (t < 256) { mall[t] = 0x7F800000u; msh[t] = 0x7F800000u; }
}

// One workgroup per batch: Gauss-Jordan inversion of SPD Sigma in LDS,
// then emit A (pair-packed for WMMA), PA (pair-packed), pAp, d_pos.
__global__ __launch_bounds__(1024) void invert_prep_kernel(
    const float* __restrict__ Z1, const float* __restrict__ Z2,
    const int* __restrict__ matching, const float* __restrict__ Sigma,
    float* __restrict__ ws)
{
    extern __shared__ float sm[];
    float* M      = sm;                    // 256*S1
    float* diff   = sm + 256 * S1;         // 16*256
    float* pAp_s  = diff + NK * DIM;       // 16
    float* dpos_s = pAp_s + 16;            // 16

    const int b   = blockIdx.x;
    const int tid = threadIdx.x;
    const float* Sg = Sigma + (size_t)b * 65536;

    for (int i = tid; i < 65536; i += 1024) {
        int d = i >> 8, e = i & 255;
        M[d * S1 + e] = Sg[i];
    }
    if (tid < 16) { pAp_s[tid] = 0.0f; dpos_s[tid] = 0.0f; }
    __syncthreads();

    // In-place Gauss-Jordan (SPD: no pivoting required).
    const int i_row = tid >> 2;
    const int j0    = (tid & 3) << 6;
    for (int k = 0; k < 256; ++k) {
        float inv = 1.0f / M[k * S1 + k];
        __syncthreads();
        if (tid < 256) M[k * S1 + tid] = (tid == k) ? 0.0f : M[k * S1 + tid] * inv;
        __syncthreads();
        float f = M[i_row * S1 + k];
        if (i_row != k) {
            const float* rk = &M[k * S1 + j0];
            float*       ri = &M[i_row * S1 + j0];
#pragma unroll 8
            for (int jj = 0; jj < 64; ++jj) ri[jj] -= f * rk[jj];
        }
        __syncthreads();
        if (tid < 256) {
            if (tid == k) M[k * S1 + k] = inv;
            else          M[tid * S1 + k] = -M[tid * S1 + k] * inv;
        }
        __syncthreads();
    }

    // Pair-packed A so a WMMA B-fragment is ONE b64 load:
    // Apk[(e>>4)*4096 + (d>>1)*32 + (e&15)*2 + (d&1)] = A[d][e]
    // -> for even d: lane l15 reads {A[d][e0+l15], A[d+1][e0+l15]} contiguously.
    float* Apk = ws + WS_APACK + (size_t)b * 65536;
    for (int i = tid; i < 65536; i += 1024) {
        int d = i >> 8, e = i & 255;
        Apk[((e >> 4) * 4096) + ((d >> 1) * 32) + ((e & 15) * 2) + (d & 1)] = M[d * S1 + e];
    }

    const float* Z1b = Z1 + (size_t)b * NK * DIM;
    const float* Z2b = Z2 + (size_t)b * NK * DIM;
    for (int i = tid; i < NK * DIM; i += 1024) {
        int k = i >> 8, e = i & 255;
        int m = matching[b * NK + k];
        diff[i] = Z1b[k * DIM + e] - Z2b[m * DIM + e];
    }
    __syncthreads();

    // PA (using A symmetric), pAp, d_pos quadratic forms.
    // PA pair-packed: PaT[(d>>1)*32 + k*2 + (d&1)] = PA[k][d]
    float* PaT = ws + WS_PAT + (size_t)b * 4096;
    for (int p = tid; p < 4096; p += 1024) {
        int k = p & 15, d = p >> 4;
        const float* Ad = &M[d * S1];
        const float* z  = &Z1b[k * DIM];
        const float* df = &diff[k * DIM];
        float pa = 0.0f, wv = 0.0f;
        for (int e = 0; e < DIM; ++e) { pa += z[e] * Ad[e]; wv += df[e] * Ad[e]; }
        PaT[((d >> 1) * 32) + (k * 2) + (d & 1)] = pa;
        atomicAdd(&pAp_s[k],  pa * z[d]);
        atomicAdd(&dpos_s[k], wv * df[d]);
    }
    __syncthreads();
    if (tid < 16) {
        ws[WS_PAP  + b * 16 + tid] = pAp_s[tid];
        ws[WS_DPOS + b * 16 + tid] = sqrtf(fmaxf(dpos_s[tid], EPSF));
    }
}

// grid (128 neg-blocks, 16 batches), 256 threads = 8 waves, 16 negatives/wave.
__global__ __launch_bounds__(256) void pairwise_min_kernel(
    const float* __restrict__ negs, float* __restrict__ ws)
{
    extern __shared__ float sm[];
    float* lneg   = sm;                   // 128 * LSTR
    float* nAn_s  = sm + 128 * LSTR;      // 128
    float* minA_s = nAn_s + 128;          // 128
    float* minS_s = minA_s + 128;         // 128

    const int b     = blockIdx.y;
    const int nbase = blockIdx.x * 128;
    const int tid   = threadIdx.x;
    const int lane  = tid & 31;
    const int w     = tid >> 5;
    const int l15   = lane & 15;
    const int half  = lane >> 4;
    const int kb    = half * 2;           // K-base for this lane half (A/B 16x4 layout)
    const int w16   = w * 16;

    // stage 128 negatives x 256 dims into padded LDS
    for (int i = tid; i < 128 * 64; i += 256) {
        int r  = i >> 6;
        int c4 = i & 63;
        float4 v = ((const float4*)(negs + (size_t)(nbase + r) * DIM))[c4];
        float* dst = &lneg[r * LSTR + c4 * 4];
        dst[0] = v.x; dst[1] = v.y; dst[2] = v.z; dst[3] = v.w;
    }
    __syncthreads();

    const float* Apk = ws + WS_APACK + (size_t)b * 65536;
    const float* PaT = ws + WS_PAT + (size_t)b * 4096;
    const float* pAp = ws + WS_PAP + b * 16;
    const float* dpo = ws + WS_DPOS + b * 16;
    unsigned int* g_mall = (unsigned int*)(ws + WS_MALL);
    unsigned int* g_msh  = (unsigned int*)(ws + WS_MSH);

    // ---- phase A: nAn[n] = n'An for this wave's 16 negatives.
    // 4 e-tiles per pass: one ds b64 A-fragment feeds 4 WMMAs; B-fragments are single b64 loads.
    float red[8] = {0, 0, 0, 0, 0, 0, 0, 0};
    for (int e0 = 0; e0 < DIM; e0 += 64) {
        v8f acc[4];
#pragma unroll
        for (int t = 0; t < 4; ++t) acc[t] = (v8f){0, 0, 0, 0, 0, 0, 0, 0};
        const float* bb = Apk + (e0 >> 4) * 4096 + l15 * 2;
#pragma unroll 2
        for (int d0 = 0; d0 < DIM; d0 += 4) {
            v2f a = *(const v2f*)&lneg[(w16 + l15) * LSTR + d0 + kb]; // A: N rows (M=n,K=d)
            const int prow = ((d0 + kb) >> 1) * 32;
#pragma unroll
            for (int t = 0; t < 4; ++t) {
                v2f bf = *(const v2f*)&bb[t * 4096 + prow];           // B: {A[d][e],A[d+1][e]}
                acc[t] = __builtin_amdgcn_wmma_f32_16x16x4_f32(
                    false, a, false, bf, (short)0, acc[t], false, false);
            }
        }
        const int rrow = (w16 + 8 * half) * LSTR + e0 + l15;
#pragma unroll
        for (int r = 0; r < 8; ++r) {                                 // fuse diag(NA . N^T)
            const int base = rrow + r * LSTR;
            red[r] += acc[0][r] * lneg[base]
                    + acc[1][r] * lneg[base + 16]
                    + acc[2][r] * lneg[base + 32]
                    + acc[3][r] * lneg[base + 48];
        }
    }
#pragma unroll
    for (int r = 0; r < 8; ++r) {
        red[r] += __shfl_xor(red[r], 1, 32);
        red[r] += __shfl_xor(red[r], 2, 32);
        red[r] += __shfl_xor(red[r], 4, 32);
        red[r] += __shfl_xor(red[r], 8, 32);
    }
    if (l15 == 0) {
#pragma unroll
        for (int r = 0; r < 8; ++r) nAn_s[w16 + 8 * half + r] = red[r];
    }
    __syncthreads();

    // ---- phase B: pAn tile (16 k x 16 n), distances, semi-hard mins
    v8f accP = {0, 0, 0, 0, 0, 0, 0, 0};
#pragma unroll 4
    for (int d0 = 0; d0 < DIM; d0 += 4) {
        v2f a  = *(const v2f*)&PaT[((d0 + kb) >> 1) * 32 + l15 * 2];   // A: PA (M=k,K=d), one b64
        v2f bf = *(const v2f*)&lneg[(w16 + l15) * LSTR + d0 + kb];     // B: N^T (K=d,N=n)
        accP = __builtin_amdgcn_wmma_f32_16x16x4_f32(
            false, a, false, bf, (short)0, accP, false, false);
    }
    float nv = nAn_s[w16 + l15];
    const float INF = __builtin_inff();
#pragma unroll
    for (int r = 0; r < 8; ++r) {
        int   k  = r + 8 * half;
        float d2 = pAp[k] - 2.0f * accP[r] + nv;
        float dd = sqrtf(fmaxf(d2, EPSF));
        float dp = dpo[k];
        float mS = (dd > dp) ? dd : INF;
        float mA = dd;
        mA = fminf(mA, __shfl_xor(mA, 1, 32)); mS = fminf(mS, __shfl_xor(mS, 1, 32));
        mA = fminf(mA, __shfl_xor(mA, 2, 32)); mS = fminf(mS, __shfl_xor(mS, 2, 32));
        mA = fminf(mA, __shfl_xor(mA, 4, 32)); mS = fminf(mS, __shfl_xor(mS, 4, 32));
        mA = fminf(mA, __shfl_xor(mA, 8, 32)); mS = fminf(mS, __shfl_xor(mS, 8, 32));
        if (l15 == 0) { minA_s[w16 + k] = mA; minS_s[w16 + k] = mS; }
    }
    __syncthreads();

    if (tid < 16) {
        float mA = minA_s[tid], mS = minS_s[tid];
        for (int ww = 1; ww < 8; ++ww) {
            mA = fminf(mA, minA_s[ww * 16 + tid]);
            mS = fminf(mS, minS_s[ww * 16 + tid]);
        }
        // non-negative floats: u32 atomic min preserves float ordering
        atomicMin(&g_mall[b * 16 + tid], __float_as_uint(mA));
        atomicMin(&g_msh [b * 16 + tid], __float_as_uint(mS));
    }
}

__global__ __launch_bounds__(256) void finish_kernel(const float* __restrict__ ws,
                                                     float* __restrict__ out)
{
    __shared__ float red[256];
    int t = threadIdx.x;
    const unsigned int* mall = (const unsigned int*)(ws + WS_MALL);
    const unsigned int* msh  = (const unsigned int*)(ws + WS_MSH);
    float dp = ws[WS_DPOS + t];
    float mv = __uint_as_float(msh[t]);
    float ma = __uint_as_float(mall[t]);
    float dn = (mv < __builtin_inff()) ? mv : ma;
    red[t] = fmaxf(dp - dn + ALPHA, 0.0f);
    __syncthreads();
    for (int s = 128; s > 0; s >>= 1) {
        if (t < s) red[t] += red[t + s];
        __syncthreads();
    }
    if (t == 0) out[0] = red[0] * (1.0f / 256.0f);
}

extern "C" void kernel_launch(void* const* d_in, const int* in_sizes, int n_in,
                              void* d_out, int out_size, void* d_ws, size_t ws_size,
                              hipStream_t stream) {
    (void)in_sizes; (void)n_in; (void)out_size; (void)ws_size;
    const float* Z1       = (const float*)d_in[0];
    const float* Z2       = (const float*)d_in[1];
    const int*   matching = (const int*)d_in[2];
    const float* negs     = (const float*)d_in[3];
    const float* Sigma    = (const float*)d_in[4];
    float* ws  = (float*)d_ws;
    float* out = (float*)d_out;

    unsigned int* mall = (unsigned int*)(ws + WS_MALL);
    unsigned int* msh  = (unsigned int*)(ws + WS_MSH);

    init_min_kernel<<<1, 256, 0, stream>>>(mall, msh);

    size_t smem1 = (size_t)(256 * S1 + NK * DIM + 32) * sizeof(float);   // ~273 KB
    invert_prep_kernel<<<16, 1024, smem1, stream>>>(Z1, Z2, matching, Sigma, ws);

    size_t smem2 = (size_t)(128 * LSTR + 3 * 128) * sizeof(float);       // ~135 KB
    pairwise_min_kernel<<<dim3(128, 16), 256, smem2, stream>>>(negs, ws);

    finish_kernel<<<1, 256, 0, stream>>>(ws, out);
}